// DistributedFNOBlock_17394617548931
// MI455X (gfx1250) — compile-verified
//
#include <hip/hip_runtime.h>

// Problem constants (from reference)
#define B_   8
#define C_   128
#define H_   256
#define W_   256
#define HW_  65536
#define M0_  32
#define M1_  32
#define NKY  64          // kept ky rows: 0..31 and 224..255 (compressed 0..63)
#define PI2  6.283185307179586f

typedef float v2f __attribute__((ext_vector_type(2)));
typedef float v8f __attribute__((ext_vector_type(8)));
typedef unsigned int u32x4 __attribute__((ext_vector_type(4)));
typedef int i32x4 __attribute__((ext_vector_type(4)));
typedef int i32x8 __attribute__((ext_vector_type(8)));

__device__ __forceinline__ v8f wmma_f32(v2f a, v2f b, v8f c) {
  // D = A(16x4 f32) * B(4x16 f32) + C(16x16 f32)
  return __builtin_amdgcn_wmma_f32_16x16x4_f32(false, a, false, b, (short)0, c, false, false);
}

__device__ __forceinline__ v8f v8f_zero() {
  v8f z = {0.f, 0.f, 0.f, 0.f, 0.f, 0.f, 0.f, 0.f};
  return z;
}

// Twiddle helper: e^{sign * i*2pi*n/256}; n reduced mod 256 for accuracy.
__device__ __forceinline__ void twiddle(int prod, float sign, float* s, float* c) {
  const int n = prod & 255;
  __sincosf(sign * (PI2 / 256.f) * (float)n, s, c);
}

// --- CDNA5 async Global->LDS copy (ASYNCcnt path) ---
__device__ __forceinline__ void async_ld_b128(unsigned lds, const float* g) {
  asm volatile("global_load_async_to_lds_b128 %0, %1, off" :: "v"(lds), "v"(g) : "memory");
}
__device__ __forceinline__ void wait_async0() {
  asm volatile("s_wait_asynccnt 0x0" ::: "memory");
}
__device__ __forceinline__ void wait_async1() {
  asm volatile("s_wait_asynccnt 0x1" ::: "memory");
}

// --- CDNA5 Tensor Data Mover: 1-D contiguous tile Global->LDS (TENSORcnt) ---
// D# built per ISA cdna5 ch.8: group0 {count,lds_addr,global_addr,type=2},
// group1 {data_size=4B, tensor_dim0=tile_dim0=nelem, tensor_dim1=1}.
// This toolchain exposes the 6-arg builtin (clang-23 / therock-10.0 headers).
__device__ __forceinline__ void tdm_load_1d(unsigned lds_addr, const void* gptr,
                                            unsigned nelem /* f32 elements, <= 65535 */) {
  const unsigned long long ga = (unsigned long long)(size_t)gptr;
  u32x4 g0;
  g0[0] = 1u;                                              // count = 1 (valid)
  g0[1] = lds_addr;                                        // lds_addr [63:32]
  g0[2] = (unsigned)ga;                                    // global_addr lo
  g0[3] = (unsigned)((ga >> 32) & 0x01FFFFFFu) | (2u << 30); // addr hi | type=2
  i32x8 g1;
  g1[0] = (int)(2u << 16);                                 // data_size = 2 (4 bytes)
  g1[1] = (int)((nelem & 0xFFFFu) << 16);                  // tensor_dim0[15:0] @ bits 63:48
  g1[2] = (int)(((nelem >> 16) & 0xFFFFu) | (1u << 16));   // tensor_dim0 hi | tensor_dim1=1
  g1[3] = (int)((nelem & 0xFFFFu) << 16);                  // tile_dim0 @ bits 127:112
  g1[4] = 0;                                               // tile_dim1=0, tile_dim2=0
  g1[5] = (int)nelem;                                      // tensor_dim0_stride lo
  g1[6] = 0;
  g1[7] = 0;
  i32x4 z4 = {0, 0, 0, 0};
  i32x8 z8 = {0, 0, 0, 0, 0, 0, 0, 0};
  __builtin_amdgcn_tensor_load_to_lds(g0, g1, z4, z4, z8, 0);
}

// ---------------------------------------------------------------------------
// Kernel 1: y0[b,o,hw] = sum_i W_lin[o,i] * x[b,i,hw]   (writes d_out)
// Per block: full M=128 (o), N=256 spatial, K=128. x K-slabs double-buffered
// in LDS via async-to-LDS (prefetch overlapped with the 16 WMMAs per slab).
// ---------------------------------------------------------------------------
__global__ __launch_bounds__(256)
void k_channel_mix(const float* __restrict__ x, const float* __restrict__ Wl,
                   float* __restrict__ out) {
  __shared__ float sx[2][4 * 256];  // double-buffered 4 K-rows x 256 spatial
  const int n0 = blockIdx.x * 256;
  const int b  = blockIdx.y;
  const int t  = threadIdx.x;
  const int lane = t & 31, wv = t >> 5;
  const int l16 = lane & 15, kh = lane >> 4;

  const float* xb = x   + (size_t)b * C_ * HW_;
  float*       ob = out + (size_t)b * C_ * HW_;

  v8f acc[16];
#pragma unroll
  for (int i = 0; i < 16; ++i) acc[i] = v8f_zero();

  const int r  = t >> 6;            // staging row 0..3
  const int c4 = (t & 63) * 4;      // staging col (float4 granularity)
  const unsigned lds0 = (unsigned)(size_t)(&sx[0][r * 256 + c4]);
  const unsigned lds1 = (unsigned)(size_t)(&sx[1][r * 256 + c4]);
  const float* gbase = xb + (size_t)r * HW_ + n0 + c4;

  // Prologue: start DMA of slab k=0 into buffer 0
  async_ld_b128(lds0, gbase);

  int p = 0;
  for (int k = 0; k < C_; k += 4) {
    if (k + 4 < C_) {
      async_ld_b128(p ? lds0 : lds1, gbase + (size_t)(k + 4) * HW_); // prefetch next slab
      wait_async1();                                                 // current slab landed
    } else {
      wait_async0();
    }
    __syncthreads();

    // A fragment: W_lin[o][k .. k+3], M = o rows
    const int o = wv * 16 + l16;
    const float2 av = *(const float2*)(Wl + o * C_ + k + 2 * kh);
    v2f a; a.x = av.x; a.y = av.y;

    const float* sp = &sx[p][0];
#pragma unroll
    for (int nt = 0; nt < 16; ++nt) {
      v2f bf;
      bf.x = sp[(2 * kh)     * 256 + nt * 16 + l16];
      bf.y = sp[(2 * kh + 1) * 256 + nt * 16 + l16];
      acc[nt] = wmma_f32(a, bf, acc[nt]);
    }
    __syncthreads();  // finish reads before the next prefetch can overwrite
    p ^= 1;
  }

#pragma unroll
  for (int nt = 0; nt < 16; ++nt) {
#pragma unroll
    for (int v = 0; v < 8; ++v) {
      const int m = wv * 16 + v + 8 * kh;
      ob[(size_t)m * HW_ + n0 + nt * 16 + l16] = acc[nt][v];
    }
  }
}

// ---------------------------------------------------------------------------
// Kernel 2: fused pruned rfft2 for one (b,c) image.
// Phase 1: T1[h,kx] = sum_w x[h,w] e^{-i2pi w kx/256}, kx<32   (WMMA, -> LDS)
// Phase 2: xf[kyc,kx] = sum_h T1[h,kx] e^{-i2pi h ky/256}, kyc compressed 0..63
// ---------------------------------------------------------------------------
__global__ __launch_bounds__(256)
void k_forward_dft(const float* __restrict__ x,
                   float* __restrict__ xfr, float* __restrict__ xfi) {
  __shared__ float s_tr[H_ * M1_];  // 32KB
  __shared__ float s_ti[H_ * M1_];  // 32KB
  const int img = blockIdx.x;       // b*C + c
  const int t = threadIdx.x;
  const int lane = t & 31, wv = t >> 5;
  const int l16 = lane & 15, kh = lane >> 4;
  const float* xi = x + (size_t)img * HW_;

  // ---- Phase 1: row-wise pruned rfft (K = w) ----
  {
    v8f aR[2][2], aI[2][2];
#pragma unroll
    for (int i = 0; i < 2; ++i)
#pragma unroll
      for (int j = 0; j < 2; ++j) { aR[i][j] = v8f_zero(); aI[i][j] = v8f_zero(); }

    const int hbase = wv * 32;
    for (int w = 0; w < W_; w += 4) {
      v2f af[2];
#pragma unroll
      for (int mt = 0; mt < 2; ++mt) {
        const int h = hbase + mt * 16 + l16;
        const float2 av = *(const float2*)(xi + (size_t)h * W_ + w + 2 * kh);
        af[mt].x = av.x; af[mt].y = av.y;
      }
#pragma unroll
      for (int nt = 0; nt < 2; ++nt) {
        const int kx = nt * 16 + l16;
        const int w0 = w + 2 * kh;
        float s0, c0, s1, c1;
        twiddle(w0 * kx, -1.f, &s0, &c0);
        twiddle((w0 + 1) * kx, -1.f, &s1, &c1);
        v2f bc; bc.x = c0; bc.y = c1;
        v2f bs; bs.x = s0; bs.y = s1;
#pragma unroll
        for (int mt = 0; mt < 2; ++mt) {
          aR[mt][nt] = wmma_f32(af[mt], bc, aR[mt][nt]);
          aI[mt][nt] = wmma_f32(af[mt], bs, aI[mt][nt]);
        }
      }
    }
#pragma unroll
    for (int mt = 0; mt < 2; ++mt)
#pragma unroll
      for (int nt = 0; nt < 2; ++nt)
#pragma unroll
        for (int v = 0; v < 8; ++v) {
          const int m = hbase + mt * 16 + v + 8 * kh;
          const int col = nt * 16 + l16;
          s_tr[m * M1_ + col] = aR[mt][nt][v];
          s_ti[m * M1_ + col] = aI[mt][nt][v];
        }
  }
  __syncthreads();

  // ---- Phase 2: column DFT on kept rows (K = h, M = kyc, N = kx) ----
  {
    const int mt = wv >> 1, nt = wv & 1;   // 4 ky-tiles x 2 kx-tiles = 8 waves
    v8f accR = v8f_zero(), accI = v8f_zero();
    const int kyc = mt * 16 + l16;                      // compressed row
    const int kya = (kyc < 32) ? kyc : (kyc + 192);     // actual ky

    for (int h = 0; h < H_; h += 4) {
      const int h0 = h + 2 * kh;
      float s0, c0, s1, c1;
      twiddle(h0 * kya, -1.f, &s0, &c0);
      twiddle((h0 + 1) * kya, -1.f, &s1, &c1);
      v2f ac;  ac.x  = c0;  ac.y  = c1;
      v2f as_; as_.x = s0;  as_.y = s1;
      v2f ans; ans.x = -s0; ans.y = -s1;

      const int col = nt * 16 + l16;
      v2f br, bi;
      br.x = s_tr[h0 * M1_ + col];       br.y = s_tr[(h0 + 1) * M1_ + col];
      bi.x = s_ti[h0 * M1_ + col];       bi.y = s_ti[(h0 + 1) * M1_ + col];

      accR = wmma_f32(ac,  br, accR);    // + c*Tr
      accR = wmma_f32(ans, bi, accR);    // - s*Ti
      accI = wmma_f32(as_, br, accI);    // + s*Tr
      accI = wmma_f32(ac,  bi, accI);    // + c*Ti
    }
#pragma unroll
    for (int v = 0; v < 8; ++v) {
      const int m = mt * 16 + v + 8 * kh;
      const int col = nt * 16 + l16;
      xfr[(size_t)img * (NKY * M1_) + m * M1_ + col] = accR[v];
      xfi[(size_t)img * (NKY * M1_) + m * M1_ + col] = accI[v];
    }
  }
}

// ---------------------------------------------------------------------------
// Kernel 3: spectral mix per mode: yf[b,o,ky,kx] = sum_i xf[b,i,ky,kx]*w[i,o,ky,kx]
// Lanes span the contiguous kx axis -> fully coalesced single pass over 256MB
// of weights (memory-bound; M=8 too small for WMMA to pay off here).
// global_prefetch_b8 issued one i ahead on the weight streams.
// ---------------------------------------------------------------------------
__global__ __launch_bounds__(256)
void k_spectral_mix(const float* __restrict__ xfr, const float* __restrict__ xfi,
                    const float* __restrict__ w1r, const float* __restrict__ w1i,
                    const float* __restrict__ w2r, const float* __restrict__ w2i,
                    float* __restrict__ yfr, float* __restrict__ yfi) {
  const int ky = blockIdx.x;            // compressed 0..63
  const int o  = blockIdx.y * 8 + (threadIdx.x >> 5);
  const int kx = threadIdx.x & 31;
  const bool lo = (ky < 32);
  const int kyw = lo ? ky : (ky - 32);
  const float* wr = lo ? w1r : w2r;
  const float* wi = lo ? w1i : w2i;

  float accR[B_], accI[B_];
#pragma unroll
  for (int b = 0; b < B_; ++b) { accR[b] = 0.f; accI[b] = 0.f; }

  const size_t wstep = (size_t)C_ * M0_ * M1_;   // i-stride in weight arrays
  for (int i = 0; i < C_; ++i) {
    const size_t widx = (((size_t)i * C_ + o) * M0_ + kyw) * M1_ + kx;
    if (i + 1 < C_) {                           // prefetch next i's weight lines
      __builtin_prefetch(&wr[widx + wstep], 0, 0);
      __builtin_prefetch(&wi[widx + wstep], 0, 0);
    }
    const float wwr = wr[widx], wwi = wi[widx];
#pragma unroll
    for (int b = 0; b < B_; ++b) {
      const size_t xidx = (((size_t)b * C_ + i) * NKY + ky) * M1_ + kx;
      const float xr = xfr[xidx], xii = xfi[xidx];
      accR[b] += xr * wwr - xii * wwi;
      accI[b] += xr * wwi + xii * wwr;
    }
  }
#pragma unroll
  for (int b = 0; b < B_; ++b) {
    const size_t yidx = (((size_t)b * C_ + o) * NKY + ky) * M1_ + kx;
    yfr[yidx] = accR[b];
    yfi[yidx] = accI[b];
  }
}

// ---------------------------------------------------------------------------
// Kernel 4: fused inverse for one (b,o) image; out += irfft2(yf).
// yf tile (2 x 8KB) is DMA'd into LDS once by the Tensor Data Mover (TENSORcnt),
// then:
// Phase 1: z[h,kx] = sum_j yf[j,kx] e^{+i2pi h ky_j/256}      (WMMA, -> LDS)
// Phase 2: ys[h,w] = (1/HW)*( Re z[h,0] + 2*sum_{kx>=1} Re(z[h,kx] e^{+i2pi w kx/256}) )
//          scale and 2x folded into twiddle B operand; out += ys.
// The yf staging aliases the (later written) z region, so static LDS stays 64KB.
// ---------------------------------------------------------------------------
__global__ __launch_bounds__(256)
void k_inverse(const float* __restrict__ yfr, const float* __restrict__ yfi,
               float* __restrict__ out) {
  __shared__ float s_zr[H_ * M1_];  // 32KB
  __shared__ float s_zi[H_ * M1_];  // 32KB
  float* s_yr = s_zr;               // alias: yf staging lives here during phase 1
  float* s_yi = s_zr + (NKY * M1_);

  const int img = blockIdx.x;
  const int t = threadIdx.x;
  const int lane = t & 31, wv = t >> 5;
  const int l16 = lane & 15, kh = lane >> 4;
  const int hbase = wv * 32;

  // ---- TDM: stage yf[img] (2048 f32 per plane) into LDS ----
  if (t < 32) {   // wave 0 issues; TDM ignores EXEC, ordered per wave
    const float* yrg = yfr + (size_t)img * (NKY * M1_);
    const float* yig = yfi + (size_t)img * (NKY * M1_);
    tdm_load_1d((unsigned)(size_t)&s_yr[0], yrg, NKY * M1_);
    tdm_load_1d((unsigned)(size_t)&s_yi[0], yig, NKY * M1_);
    __builtin_amdgcn_s_wait_tensorcnt(0);
  }
  __syncthreads();

  // ---- Phase 1: inverse row transform (K = j over 64 kept ky) ----
  v8f zR[2][2], zI[2][2];
#pragma unroll
  for (int i = 0; i < 2; ++i)
#pragma unroll
    for (int j = 0; j < 2; ++j) { zR[i][j] = v8f_zero(); zI[i][j] = v8f_zero(); }

  for (int j = 0; j < NKY; j += 4) {
    const int j0 = j + 2 * kh;
    const int kyA0 = (j0     < 32) ? j0       : (j0 + 192);
    const int kyA1 = (j0 + 1 < 32) ? (j0 + 1) : (j0 + 193);

    v2f ac[2], as_[2], ans[2];
#pragma unroll
    for (int mt = 0; mt < 2; ++mt) {
      const int h = hbase + mt * 16 + l16;
      float s0, c0, s1, c1;
      twiddle(h * kyA0, 1.f, &s0, &c0);
      twiddle(h * kyA1, 1.f, &s1, &c1);
      ac[mt].x  = c0;  ac[mt].y  = c1;
      as_[mt].x = s0;  as_[mt].y = s1;
      ans[mt].x = -s0; ans[mt].y = -s1;
    }
#pragma unroll
    for (int nt = 0; nt < 2; ++nt) {
      const int col = nt * 16 + l16;
      v2f br, bi;
      br.x = s_yr[j0 * M1_ + col];       br.y = s_yr[(j0 + 1) * M1_ + col];
      bi.x = s_yi[j0 * M1_ + col];       bi.y = s_yi[(j0 + 1) * M1_ + col];
#pragma unroll
      for (int mt = 0; mt < 2; ++mt) {
        zR[mt][nt] = wmma_f32(ac[mt],  br, zR[mt][nt]);   // + c*yr
        zR[mt][nt] = wmma_f32(ans[mt], bi, zR[mt][nt]);   // - s*yi
        zI[mt][nt] = wmma_f32(as_[mt], br, zI[mt][nt]);   // + s*yr
        zI[mt][nt] = wmma_f32(ac[mt],  bi, zI[mt][nt]);   // + c*yi
      }
    }
  }
  __syncthreads();   // all yf reads done before z overwrites the aliased region

#pragma unroll
  for (int mt = 0; mt < 2; ++mt)
#pragma unroll
    for (int nt = 0; nt < 2; ++nt)
#pragma unroll
      for (int v = 0; v < 8; ++v) {
        const int m = hbase + mt * 16 + v + 8 * kh;
        const int col = nt * 16 + l16;
        s_zr[m * M1_ + col] = zR[mt][nt][v];
        s_zi[m * M1_ + col] = zI[mt][nt][v];
      }
  __syncthreads();

  // ---- Phase 2: inverse column transform + accumulate into out ----
  {
    const float inv = 1.0f / ((float)H_ * (float)W_);
    float* ob = out + (size_t)img * HW_;

    for (int nt = 0; nt < 16; ++nt) {
      v8f acc[2];
      acc[0] = v8f_zero(); acc[1] = v8f_zero();
      const int wcol = nt * 16 + l16;

      for (int k = 0; k < M1_; k += 4) {
        const int k0 = k + 2 * kh;
        float s0, c0, s1, c1;
        twiddle(wcol * k0, 1.f, &s0, &c0);
        twiddle(wcol * (k0 + 1), 1.f, &s1, &c1);
        const float f0 = (k0 == 0) ? inv : (2.f * inv);
        const float f1 = 2.f * inv;              // k0+1 >= 1 always
        v2f bc;  bc.x  = f0 * c0;   bc.y  = f1 * c1;
        v2f bns; bns.x = -f0 * s0;  bns.y = -f1 * s1;

#pragma unroll
        for (int mt = 0; mt < 2; ++mt) {
          const int h = hbase + mt * 16 + l16;
          const float2 arv = *(const float2*)&s_zr[h * M1_ + k0];
          const float2 aiv = *(const float2*)&s_zi[h * M1_ + k0];
          v2f ar; ar.x = arv.x; ar.y = arv.y;
          v2f ai; ai.x = aiv.x; ai.y = aiv.y;
          acc[mt] = wmma_f32(ar, bc,  acc[mt]);   // + zr * coef*cos
          acc[mt] = wmma_f32(ai, bns, acc[mt]);   // - zi * coef*sin
        }
      }
#pragma unroll
      for (int mt = 0; mt < 2; ++mt)
#pragma unroll
        for (int v = 0; v < 8; ++v) {
          const int h = hbase + mt * 16 + v + 8 * kh;
          ob[(size_t)h * W_ + wcol] += acc[mt][v];
        }
    }
  }
}

// ---------------------------------------------------------------------------
extern "C" void kernel_launch(void* const* d_in, const int* in_sizes, int n_in,
                              void* d_out, int out_size, void* d_ws, size_t ws_size,
                              hipStream_t stream) {
  const float* x   = (const float*)d_in[0];
  const float* Wl  = (const float*)d_in[1];
  const float* w1r = (const float*)d_in[2];
  const float* w1i = (const float*)d_in[3];
  const float* w2r = (const float*)d_in[4];
  const float* w2i = (const float*)d_in[5];
  float* out = (float*)d_out;

  // Workspace: xf / yf complex planes, each B*C*NKY*M1 = 2M floats (8MB) -> 32MB total
  const size_t plane = (size_t)B_ * C_ * NKY * M1_;
  float* xfr = (float*)d_ws;
  float* xfi = xfr + plane;
  float* yfr = xfi + plane;
  float* yfi = yfr + plane;

  // 1) y0 = W_lin * x  -> out
  k_channel_mix<<<dim3(HW_ / 256, B_), 256, 0, stream>>>(x, Wl, out);
  // 2) pruned rfft2 -> xf
  k_forward_dft<<<dim3(B_ * C_), 256, 0, stream>>>(x, xfr, xfi);
  // 3) per-mode channel mix -> yf
  k_spectral_mix<<<dim3(NKY, C_ / 8), 256, 0, stream>>>(xfr, xfi, w1r, w1i, w2r, w2i, yfr, yfi);
  // 4) out += irfft2(yf)
  k_inverse<<<dim3(B_ * C_), 256, 0, stream>>>(yfr, yfi, out);
}